// GRU_61203283968688
// MI455X (gfx1250) — compile-verified
//
#include <hip/hip_runtime.h>

// MI455X / gfx1250 fused 3-layer GRU + MLP head.
// Grid: 32 workgroups x 128 threads (4 wave32). Each WG owns 16 batch rows
// and advances all 3 GRU layers through all 512 timesteps (stacked-GRU
// per-timestep fusion), keeping all weights resident in LDS (~278 KB of the
// 320 KB/WGP CDNA5 budget) and all gate math in WMMA f32 accumulators.

typedef __attribute__((ext_vector_type(2))) float v2f;
typedef __attribute__((ext_vector_type(8))) float v8f;

namespace {
constexpr int kT   = 512;
constexpr int kDin = 22;
constexpr int kH   = 64;
constexpr int kRows = 16;     // batch rows per WG == WMMA M
constexpr int kThreads = 128; // 4 waves: wave w owns gate columns [16w,16w+16)

// LDS layout (float offsets)
constexpr int OFF_WIH0 = 0;                    // [192][24] (k padded 22->24)
constexpr int OFF_WHH0 = OFF_WIH0 + 192 * 24;  // [192][64]
constexpr int OFF_WIH1 = OFF_WHH0 + 192 * 64;  // [192][64]
constexpr int OFF_WHH1 = OFF_WIH1 + 192 * 64;
constexpr int OFF_WIH2 = OFF_WHH1 + 192 * 64;
constexpr int OFF_WHH2 = OFF_WIH2 + 192 * 64;
constexpr int OFF_H0   = OFF_WHH2 + 192 * 64;  // [16][64]
constexpr int OFF_H1   = OFF_H0 + 16 * 64;
constexpr int OFF_H2   = OFF_H1 + 16 * 64;
constexpr int OFF_XS   = OFF_H2 + 16 * 64;     // [16][24]
constexpr int OFF_Y1   = OFF_XS + 16 * 24;     // [16][56]
constexpr int OFF_Y2   = OFF_Y1 + 16 * 56;     // [16][44]
constexpr int kSmemFloats = OFF_Y2 + 16 * 44;  // 71104 floats = 284416 B
} // namespace

__device__ __forceinline__ v8f wmma4(v2f a, v2f b, v8f c) {
  // D = A(16x4 f32) * B(4x16 f32) + C(16x16 f32)
  return __builtin_amdgcn_wmma_f32_16x16x4_f32(
      /*neg_a=*/false, a, /*neg_b=*/false, b,
      /*c_mod=*/(short)0, c, /*reuse_a=*/false, /*reuse_b=*/false);
}

__device__ __forceinline__ v2f ld2(const float* p) {
  return *(const v2f*)p;  // 8B-aligned by construction -> ds_load_b64
}

__device__ __forceinline__ float fsig(float x) {
  return 1.0f / (1.0f + __expf(-x));
}
__device__ __forceinline__ float ftanh(float x) {
  float e = __expf(-2.0f * fabsf(x));
  float r = (1.0f - e) / (1.0f + e);
  return copysignf(r, x);
}

// One GRU layer for one timestep:
//   gi = A_gi * W_ihT (+b_ih), gh = h * W_hhT (+b_hh), gates, h update.
// Wave w's lanes own gate columns jcol = 16w + (lane&15); accumulator tile
// element e lives at (row crow+e, col jcol) per the CDNA5 C/D layout.
template <int KCGI>
__device__ __forceinline__ void layer_phase(
    const float* __restrict__ agi, int lda,     // gi A-matrix [16][lda] in LDS
    const float* __restrict__ wih, int ldbi,    // [192][ldbi] in LDS
    float* __restrict__ hlds,                   // [16][64] in LDS (old h in, new h out)
    const float* __restrict__ whh,              // [192][64] in LDS
    float br, float bz, float bni, float bnh,   // per-lane bias scalars
    float* hreg,                                // [8] this lane's h tile
    int lm, int kb, int jcol, int crow)
{
  v8f racc, zacc, niacc, nhacc;
#pragma unroll
  for (int e = 0; e < 8; ++e) {
    racc[e] = br; zacc[e] = bz; niacc[e] = bni; nhacc[e] = bnh;
  }

  // gi GEMM: A row = lm, K pattern per lane half (kb in {0,2})
  {
    const float* arow = agi + lm * lda;
    const float* wr = wih + (0   + jcol) * ldbi;
    const float* wz = wih + (64  + jcol) * ldbi;
    const float* wn = wih + (128 + jcol) * ldbi;
#pragma unroll
    for (int kc = 0; kc < KCGI; ++kc) {
      const int ko = kc * 4 + kb;
      v2f a = ld2(arow + ko);
      racc  = wmma4(a, ld2(wr + ko), racc);
      zacc  = wmma4(a, ld2(wz + ko), zacc);
      niacc = wmma4(a, ld2(wn + ko), niacc);
    }
  }
  // gh GEMM (K = 64 -> 16 chunks)
  {
    const float* hrow = hlds + lm * kH;
    const float* ur = whh + (0   + jcol) * kH;
    const float* uz = whh + (64  + jcol) * kH;
    const float* un = whh + (128 + jcol) * kH;
#pragma unroll
    for (int kc = 0; kc < 16; ++kc) {
      const int ko = kc * 4 + kb;
      v2f a = ld2(hrow + ko);
      racc  = wmma4(a, ld2(ur + ko), racc);
      zacc  = wmma4(a, ld2(uz + ko), zacc);
      nhacc = wmma4(a, ld2(un + ko), nhacc);
    }
  }

  // Gates entirely in registers (this wave owns the full r/z/n triplet
  // for its 16 columns).
  float hn[8];
#pragma unroll
  for (int e = 0; e < 8; ++e) {
    float r = fsig(racc[e]);
    float z = fsig(zacc[e]);
    float n = ftanh(niacc[e] + r * nhacc[e]);
    hn[e] = (1.0f - z) * n + z * hreg[e];
    hreg[e] = hn[e];
  }
  __syncthreads();  // all waves done reading old h before anyone writes
#pragma unroll
  for (int e = 0; e < 8; ++e) {
    hlds[(crow + e) * kH + jcol] = hn[e];
  }
  __syncthreads();  // new h visible to all waves
}

__global__ void __launch_bounds__(kThreads, 1)
gru3_fused_kernel(const float* __restrict__ x,
                  const float* __restrict__ wih0, const float* __restrict__ whh0,
                  const float* __restrict__ bih0, const float* __restrict__ bhh0,
                  const float* __restrict__ wih1, const float* __restrict__ whh1,
                  const float* __restrict__ bih1, const float* __restrict__ bhh1,
                  const float* __restrict__ wih2, const float* __restrict__ whh2,
                  const float* __restrict__ bih2, const float* __restrict__ bhh2,
                  const float* __restrict__ fc1w, const float* __restrict__ fc1b,
                  const float* __restrict__ bn1g, const float* __restrict__ bn1b,
                  const float* __restrict__ bn1m, const float* __restrict__ bn1v,
                  const float* __restrict__ fc2w, const float* __restrict__ fc2b,
                  const float* __restrict__ bn2g, const float* __restrict__ bn2b,
                  const float* __restrict__ bn2m, const float* __restrict__ bn2v,
                  const float* __restrict__ fc3w, const float* __restrict__ fc3b,
                  float* __restrict__ out)
{
  extern __shared__ float smem[];
  const int tid = threadIdx.x;
  const int b0  = blockIdx.x * kRows;

  float* s_wih0 = smem + OFF_WIH0;
  float* s_whh0 = smem + OFF_WHH0;
  float* s_wih1 = smem + OFF_WIH1;
  float* s_whh1 = smem + OFF_WHH1;
  float* s_wih2 = smem + OFF_WIH2;
  float* s_whh2 = smem + OFF_WHH2;
  float* s_h0   = smem + OFF_H0;
  float* s_h1   = smem + OFF_H1;
  float* s_h2   = smem + OFF_H2;
  float* s_xs   = smem + OFF_XS;
  float* s_y1   = smem + OFF_Y1;
  float* s_y2   = smem + OFF_Y2;

  // ---- stage weights into LDS (once) ----
  for (int i = tid; i < 192 * 24; i += kThreads) {
    int n = i / 24, k = i - n * 24;
    s_wih0[i] = (k < kDin) ? wih0[n * kDin + k] : 0.0f;
  }
  for (int i = tid; i < 192 * 64; i += kThreads) {
    s_whh0[i] = whh0[i];
    s_wih1[i] = wih1[i];
    s_whh1[i] = whh1[i];
    s_wih2[i] = wih2[i];
    s_whh2[i] = whh2[i];
  }
  for (int i = tid; i < 16 * 64; i += kThreads) {
    s_h0[i] = 0.0f; s_h1[i] = 0.0f; s_h2[i] = 0.0f;
  }

  // ---- x_t staging map: 16x22 slice, 3 elements per thread ----
  int   xbase[3], xoff[3];
  bool  xval[3];
#pragma unroll
  for (int q = 0; q < 3; ++q) {
    int i = tid + kThreads * q;
    xval[q] = (i < kRows * kDin);
    int ii = xval[q] ? i : 0;
    int m = ii / kDin, d = ii - m * kDin;
    xbase[q] = (b0 + m) * kDin * kT + d * kT;  // + t
    xoff[q]  = m * 24 + d;
  }
  // stage x_0, zero K-pad columns 22..23
#pragma unroll
  for (int q = 0; q < 3; ++q)
    if (xval[q]) s_xs[xoff[q]] = x[xbase[q]];
  if (tid < kRows) { s_xs[tid * 24 + 22] = 0.0f; s_xs[tid * 24 + 23] = 0.0f; }

  // ---- per-lane constants (CDNA5 WMMA 16x16x4 f32 operand layout) ----
  const int lane = tid & 31;
  const int wid  = tid >> 5;
  const int lm   = lane & 15;          // A row / B col within tile
  const int kb   = (lane >> 4) * 2;    // K sub-offset: lanes 0-15 -> K 0,1; 16-31 -> K 2,3
  const int jcol = wid * 16 + lm;      // gate column owned by this lane
  const int crow = (lane >> 4) * 8;    // C/D tile: VGPR e -> row crow+e

  // per-lane bias scalars (b_ih + b_hh pre-summed for r,z; split for n)
  const float br0 = bih0[jcol] + bhh0[jcol];
  const float bz0 = bih0[64 + jcol] + bhh0[64 + jcol];
  const float bni0 = bih0[128 + jcol], bnh0 = bhh0[128 + jcol];
  const float br1 = bih1[jcol] + bhh1[jcol];
  const float bz1 = bih1[64 + jcol] + bhh1[64 + jcol];
  const float bni1 = bih1[128 + jcol], bnh1 = bhh1[128 + jcol];
  const float br2 = bih2[jcol] + bhh2[jcol];
  const float bz2 = bih2[64 + jcol] + bhh2[64 + jcol];
  const float bni2 = bih2[128 + jcol], bnh2 = bhh2[128 + jcol];

  float h0r[8], h1r[8], h2r[8];
#pragma unroll
  for (int e = 0; e < 8; ++e) { h0r[e] = 0.0f; h1r[e] = 0.0f; h2r[e] = 0.0f; }

  __syncthreads();

  // ---- serial scan over T, all 3 layers fused per timestep ----
  for (int t = 0; t < kT; ++t) {
    // software-prefetch x_{t+1} into registers; latency hidden by the
    // full step of WMMA work below
    float nx[3] = {0.0f, 0.0f, 0.0f};
    const bool more = (t + 1) < kT;
    if (more) {
#pragma unroll
      for (int q = 0; q < 3; ++q)
        if (xval[q]) nx[q] = x[xbase[q] + t + 1];
    }

    layer_phase<6 >(s_xs, 24, s_wih0, 24, s_h0, s_whh0,
                    br0, bz0, bni0, bnh0, h0r, lm, kb, jcol, crow);
    layer_phase<16>(s_h0, 64, s_wih1, 64, s_h1, s_whh1,
                    br1, bz1, bni1, bnh1, h1r, lm, kb, jcol, crow);
    layer_phase<16>(s_h1, 64, s_wih2, 64, s_h2, s_whh2,
                    br2, bz2, bni2, bnh2, h2r, lm, kb, jcol, crow);

    if (more) {
#pragma unroll
      for (int q = 0; q < 3; ++q)
        if (xval[q]) s_xs[xoff[q]] = nx[q];
      __syncthreads();
    }
  }

  // ---- MLP head on h2 (16 rows): fc1+bn1+relu, fc2+bn2+relu, fc3 ----
  constexpr float kEps = 1e-5f;
  for (int i = tid; i < 16 * 54; i += kThreads) {
    int m = i / 54, o = i - m * 54;
    float acc = fc1b[o];
    for (int k = 0; k < 64; ++k) acc += s_h2[m * 64 + k] * fc1w[o * 64 + k];
    float yv = (acc - bn1m[o]) * rsqrtf(bn1v[o] + kEps) * bn1g[o] + bn1b[o];
    s_y1[m * 56 + o] = fmaxf(yv, 0.0f);
  }
  __syncthreads();
  for (int i = tid; i < 16 * 44; i += kThreads) {
    int m = i / 44, o = i - m * 44;
    float acc = fc2b[o];
    for (int k = 0; k < 54; ++k) acc += s_y1[m * 56 + k] * fc2w[o * 54 + k];
    float yv = (acc - bn2m[o]) * rsqrtf(bn2v[o] + kEps) * bn2g[o] + bn2b[o];
    s_y2[m * 44 + o] = fmaxf(yv, 0.0f);
  }
  __syncthreads();
  for (int i = tid; i < 16 * 4; i += kThreads) {
    int m = i / 4, o = i - m * 4;
    float acc = fc3b[o];
    for (int k = 0; k < 44; ++k) acc += s_y2[m * 44 + k] * fc3w[o * 44 + k];
    out[(b0 + m) * 4 + o] = acc;
  }
}

extern "C" void kernel_launch(void* const* d_in, const int* in_sizes, int n_in,
                              void* d_out, int out_size, void* d_ws, size_t ws_size,
                              hipStream_t stream) {
  (void)in_sizes; (void)n_in; (void)out_size; (void)d_ws; (void)ws_size;
  const float* P[27];
  for (int i = 0; i < 27; ++i) P[i] = (const float*)d_in[i];

  const size_t smem_bytes = (size_t)kSmemFloats * sizeof(float);  // ~278 KB
  (void)hipFuncSetAttribute((const void*)gru3_fused_kernel,
                            hipFuncAttributeMaxDynamicSharedMemorySize,
                            (int)smem_bytes);

  dim3 grid(512 / kRows);   // 32 workgroups, one 16-row batch tile each
  dim3 block(kThreads);     // 4 wave32
  gru3_fused_kernel<<<grid, block, smem_bytes, stream>>>(
      P[0],
      P[1], P[2], P[3], P[4],
      P[5], P[6], P[7], P[8],
      P[9], P[10], P[11], P[12],
      P[13], P[14], P[15], P[16], P[17], P[18],
      P[19], P[20], P[21], P[22], P[23], P[24],
      P[25], P[26],
      (float*)d_out);
}